// GIN_53609781789214
// MI455X (gfx1250) — compile-verified
//
#include <hip/hip_runtime.h>
#include <hip/hip_bf16.h>

#define N_NODES 100000
#define N_EDGES 1600000
#define HIDDEN  128
#define N_LAYERS 4
#define ROWS_PB 32                       // rows per block; 100000 % 32 == 0

typedef __attribute__((ext_vector_type(2))) float v2f;
typedef __attribute__((ext_vector_type(8))) float v8f;

// ---------------------------------------------------------------------------
// agg[i] = (1+eps) * x[i]  (eps == 0)  -- float4 vectorized copy
// ---------------------------------------------------------------------------
__global__ __launch_bounds__(256) void gin_init_agg(const float4* __restrict__ x,
                                                    float4* __restrict__ agg,
                                                    int n4) {
    int i = blockIdx.x * blockDim.x + threadIdx.x;
    if (i < n4) agg[i] = x[i];
}

// ---------------------------------------------------------------------------
// agg[dst] += x[src]  : one wave32 per edge, lane handles 4 features.
// src/dst are wave-uniform -> scalar loads; gather is a coalesced float4;
// atomics are non-returning (fire-and-forget) global_atomic_add_f32.
// ---------------------------------------------------------------------------
__global__ __launch_bounds__(256) void gin_scatter(const float* __restrict__ x,
                                                   const int* __restrict__ src,
                                                   const int* __restrict__ dst,
                                                   float* __restrict__ agg,
                                                   int n_edges) {
    int gid  = blockIdx.x * blockDim.x + threadIdx.x;
    int e    = gid >> 5;
    int lane = gid & 31;
    if (e >= n_edges) return;
    int s = src[e];
    int d = dst[e];
    const float4 v = *(const float4*)(x + (size_t)s * HIDDEN + lane * 4);
    float* ap = agg + (size_t)d * HIDDEN + lane * 4;
    atomicAdd(ap + 0, v.x);
    atomicAdd(ap + 1, v.y);
    atomicAdd(ap + 2, v.z);
    atomicAdd(ap + 3, v.w);
}

// ---------------------------------------------------------------------------
// Fused GIN MLP: xout = xin + relu( relu(agg@W1 + b1) @ W2 + b2 )
// Block = 256 threads = 8 waves; block computes rows [row0, row0+32) as two
// 16-row tiles, wave w owns the 16-wide column stripe at n0 = 16w.
// Each B (weight) fragment is reused by 2 WMMAs on independent accumulator
// chains c0/c1 -> fills the WMMA->WMMA co-exec hazard slots and halves the
// L2 weight traffic vs 16 rows/block.
// ---------------------------------------------------------------------------
__global__ __launch_bounds__(256) void gin_mlp(const float* __restrict__ agg,
                                               const float* __restrict__ xin,
                                               const float* __restrict__ W1,
                                               const float* __restrict__ b1,
                                               const float* __restrict__ W2,
                                               const float* __restrict__ b2,
                                               float* __restrict__ xout) {
    // padded to 132 floats/row: conflict-free banks for the strided b64 reads
    // (132 % 64 == 4) and the half-wave split writes
    __shared__ float Hs[ROWS_PB][HIDDEN + 4];

    const int lane  = threadIdx.x & 31;
    const int wave  = threadIdx.x >> 5;        // 0..7 -> N tile
    const int n0    = wave * 16;
    const int row0  = blockIdx.x * ROWS_PB;    // exact division, no tail
    const int lhalf = lane & 15;
    const int hi    = lane >> 4;               // 0: lanes 0-15 (K=0,1 / M=r); 1: lanes 16-31 (K=2,3 / M=r+8)

    // ---- GEMM1: C = agg[row0:row0+32, :] @ W1 ----
    v8f c0 = {}, c1 = {};
    const float* arow0 = agg + (size_t)(row0 + lhalf) * HIDDEN + hi * 2;
    const float* arow1 = arow0 + 16 * HIDDEN;
    #pragma unroll 8
    for (int k = 0; k < HIDDEN; k += 4) {
        const float* wcol = W1 + (size_t)(k + hi * 2) * HIDDEN + n0 + lhalf;
        v2f b;                                  // B frag: 4x16 f32 (2 VGPRs)
        b.x = wcol[0];
        b.y = wcol[HIDDEN];
        v2f a0, a1;                             // A frags: 16x4 f32 (2 VGPRs)
        a0.x = arow0[k + 0];  a0.y = arow0[k + 1];
        a1.x = arow1[k + 0];  a1.y = arow1[k + 1];
        c0 = __builtin_amdgcn_wmma_f32_16x16x4_f32(false, a0, false, b,
                                                   (short)0, c0, false, false);
        c1 = __builtin_amdgcn_wmma_f32_16x16x4_f32(false, a1, false, b,
                                                   (short)0, c1, false, false);
    }

    // bias + ReLU, spill to LDS (re-stripes C layout -> A layout for GEMM2)
    const float bias1 = b1[n0 + lhalf];
    #pragma unroll
    for (int r = 0; r < 8; ++r) {
        float v0 = c0[r] + bias1;
        float v1 = c1[r] + bias1;
        Hs[r + hi * 8][n0 + lhalf]      = v0 > 0.0f ? v0 : 0.0f;
        Hs[16 + r + hi * 8][n0 + lhalf] = v1 > 0.0f ? v1 : 0.0f;
    }
    __syncthreads();

    // ---- GEMM2: C2 = Hs @ W2 ----
    v8f d0 = {}, d1 = {};
    #pragma unroll 8
    for (int k = 0; k < HIDDEN; k += 4) {
        const float* wcol = W2 + (size_t)(k + hi * 2) * HIDDEN + n0 + lhalf;
        v2f b;
        b.x = wcol[0];
        b.y = wcol[HIDDEN];
        v2f a0, a1;
        a0.x = Hs[lhalf][k + hi * 2 + 0];
        a0.y = Hs[lhalf][k + hi * 2 + 1];
        a1.x = Hs[16 + lhalf][k + hi * 2 + 0];
        a1.y = Hs[16 + lhalf][k + hi * 2 + 1];
        d0 = __builtin_amdgcn_wmma_f32_16x16x4_f32(false, a0, false, b,
                                                   (short)0, d0, false, false);
        d1 = __builtin_amdgcn_wmma_f32_16x16x4_f32(false, a1, false, b,
                                                   (short)0, d1, false, false);
    }

    // bias + ReLU + residual, store
    const float bias2 = b2[n0 + lhalf];
    #pragma unroll
    for (int r = 0; r < 8; ++r) {
        {
            int row = row0 + r + hi * 8;
            float h = d0[r] + bias2;
            h = h > 0.0f ? h : 0.0f;
            size_t idx = (size_t)row * HIDDEN + n0 + lhalf;
            xout[idx] = xin[idx] + h;
        }
        {
            int row = row0 + 16 + r + hi * 8;
            float h = d1[r] + bias2;
            h = h > 0.0f ? h : 0.0f;
            size_t idx = (size_t)row * HIDDEN + n0 + lhalf;
            xout[idx] = xin[idx] + h;
        }
    }
}

// ---------------------------------------------------------------------------
extern "C" void kernel_launch(void* const* d_in, const int* in_sizes, int n_in,
                              void* d_out, int out_size, void* d_ws, size_t ws_size,
                              hipStream_t stream) {
    const float* x0  = (const float*)d_in[0];
    const int*   ei  = (const int*)d_in[1];      // edge_index [2, N_EDGES] (int32)
    const float* W1  = (const float*)d_in[2];    // [4,128,128]
    const float* b1  = (const float*)d_in[3];    // [4,128]
    const float* W2  = (const float*)d_in[4];
    const float* b2  = (const float*)d_in[5];
    float*       out = (float*)d_out;

    const int* src = ei;                 // row 0: message sources
    const int* dst = ei + N_EDGES;       // row 1: aggregation targets

    float* agg  = (float*)d_ws;                          // 51.2 MB
    float* xbuf = agg + (size_t)N_NODES * HIDDEN;        // 51.2 MB

    const int n4          = N_NODES * HIDDEN / 4;        // 3.2M float4
    const int initBlocks  = (n4 + 255) / 256;
    const long long sthreads = (long long)N_EDGES * 32;
    const int scatBlocks  = (int)((sthreads + 255) / 256);
    const int mlpBlocks   = N_NODES / ROWS_PB;           // 3125 exactly

    for (int l = 0; l < N_LAYERS; ++l) {
        const float* xin  = (l == 0) ? x0 : xbuf;
        float*       xout = (l == N_LAYERS - 1) ? out : xbuf;

        // agg = (1+eps)*x
        gin_init_agg<<<initBlocks, 256, 0, stream>>>((const float4*)xin,
                                                     (float4*)agg, n4);
        // agg[dst] += x[src]
        gin_scatter<<<scatBlocks, 256, 0, stream>>>(xin, src, dst, agg, N_EDGES);
        // xout = xin + relu(MLP(agg))
        gin_mlp<<<mlpBlocks, 256, 0, stream>>>(agg, xin,
                                               W1 + (size_t)l * HIDDEN * HIDDEN,
                                               b1 + (size_t)l * HIDDEN,
                                               W2 + (size_t)l * HIDDEN * HIDDEN,
                                               b2 + (size_t)l * HIDDEN,
                                               xout);
    }
}